// RadialAngularEmbedding_19765439496446
// MI455X (gfx1250) — compile-verified
//
#include <hip/hip_runtime.h>

#define NN 50000
#define NE 1000000
#define CC 32
#define OUTC 288          // 32 + 96 + 160
#define NTILES (NN / 16)  // 3125, exact

typedef __attribute__((ext_vector_type(2))) float v2f;
typedef __attribute__((ext_vector_type(8))) float v8f;

__device__ __forceinline__ void atom_add(float* p, float v) {
    (void)__hip_atomic_fetch_add(p, v, __ATOMIC_RELAXED, __HIP_MEMORY_SCOPE_AGENT);
}

__global__ void zero_kernel(float* __restrict__ p, int n) {
    int i = blockIdx.x * blockDim.x + threadIdx.x;
    if (i < n) p[i] = 0.0f;
}

// ---------------------------------------------------------------------------
// Edge kernel: 4 waves/block, each wave owns 32 edges.
// Phase 1: lane l runs the whole 8->6->6->6->96 MLP for edge (base+l), writes
//          96 tp values to LDS (stride 97 -> conflict-free both phases).
// Phase 2: wave iterates its 32 edges; lane l = channel l; 9 f32 atomics/lane.
// ---------------------------------------------------------------------------
#define EPW 32       // edges per wave
#define WPB 4        // waves per block
#define TPS 97       // tp LDS stride (33*l mod 64 is a permutation -> no conflicts)

__global__ __launch_bounds__(128) void edge_kernel(
    const float* __restrict__ lenght,
    const float* __restrict__ node_features,
    const float* __restrict__ edge_attr,
    const long long* __restrict__ edge_index,
    const float* __restrict__ fw0,   // (8,6)
    const float* __restrict__ fw1,   // (6,6)
    const float* __restrict__ fw2,   // (6,6)
    const float* __restrict__ fw3,   // (6,96)
    float* __restrict__ msg)         // (NN, 288) accumulator
{
    __shared__ float s_w0[48];
    __shared__ float s_w1[36];
    __shared__ float s_w2[36];
    __shared__ float s_w3[576];
    __shared__ float s_tp[WPB * EPW * TPS];

    const int tid = threadIdx.x;
    for (int i = tid; i < 48;  i += 128) s_w0[i] = fw0[i];
    for (int i = tid; i < 36;  i += 128) s_w1[i] = fw1[i];
    for (int i = tid; i < 36;  i += 128) s_w2[i] = fw2[i];
    for (int i = tid; i < 576; i += 128) s_w3[i] = fw3[i];
    __syncthreads();

    const int wave = tid >> 5;
    const int lane = tid & 31;
    const long long ebase = (long long)blockIdx.x * (WPB * EPW) + (long long)wave * EPW;

    // ---- Phase 1: per-lane edge MLP ----
    {
        const long long e = ebase + lane;
        if (e < NE) {
            const float inv8 = 0.35355339059327373f;  // 1/sqrt(8)
            const float inv6 = 0.40824829046386307f;  // 1/sqrt(6)
            float len[8];
#pragma unroll
            for (int k = 0; k < 8; ++k) len[k] = lenght[e * 8 + k];

            float h0[6], h1[6], h2[6];
#pragma unroll
            for (int j = 0; j < 6; ++j) {
                float t = 0.f;
#pragma unroll
                for (int k = 0; k < 8; ++k) t += len[k] * s_w0[k * 6 + j];
                t *= inv8;
                h0[j] = t / (1.f + __expf(-t));
            }
#pragma unroll
            for (int j = 0; j < 6; ++j) {
                float t = 0.f;
#pragma unroll
                for (int k = 0; k < 6; ++k) t += h0[k] * s_w1[k * 6 + j];
                t *= inv6;
                h1[j] = t / (1.f + __expf(-t));
            }
#pragma unroll
            for (int j = 0; j < 6; ++j) {
                float t = 0.f;
#pragma unroll
                for (int k = 0; k < 6; ++k) t += h1[k] * s_w2[k * 6 + j];
                t *= inv6;
                h2[j] = t / (1.f + __expf(-t));
            }
            float* tp = &s_tp[(wave * EPW + lane) * TPS];
            for (int m = 0; m < 96; ++m) {
                float a = 0.f;
#pragma unroll
                for (int k = 0; k < 6; ++k) a += h2[k] * s_w3[k * 96 + m];
                tp[m] = a * inv6;
            }
        }
    }
    __syncthreads();

    // ---- Phase 2: scatter messages; lane = channel ----
    for (int j = 0; j < EPW; ++j) {
        const long long e = ebase + j;
        if (e >= NE) break;
        const long long s = edge_index[e];
        const long long r = edge_index[NE + e];

        const float x = node_features[s * CC + lane];      // coalesced 128B
        const float v = edge_attr[e * 9 + (lane < 9 ? lane : 0)];
        const float y0  = __shfl(v, 0, 32);
        const float y1a = __shfl(v, 1, 32), y1b = __shfl(v, 2, 32), y1c = __shfl(v, 3, 32);
        const float y2a = __shfl(v, 4, 32), y2b = __shfl(v, 5, 32), y2c = __shfl(v, 6, 32),
                    y2d = __shfl(v, 7, 32), y2e = __shfl(v, 8, 32);

        const float* tp = &s_tp[(wave * EPW + j) * TPS];
        const float wx0 = tp[lane]      * x;
        const float wx1 = tp[32 + lane] * x;
        const float wx2 = tp[64 + lane] * x;

        float* out = msg + (long long)r * OUTC;
        atom_add(out + lane, wx0 * y0);
        float* o1 = out + 32 + lane * 3;
        atom_add(o1 + 0, wx1 * y1a);
        atom_add(o1 + 1, wx1 * y1b);
        atom_add(o1 + 2, wx1 * y1c);
        float* o2 = out + 128 + lane * 5;
        atom_add(o2 + 0, wx2 * y2a);
        atom_add(o2 + 1, wx2 * y2b);
        atom_add(o2 + 2, wx2 * y2c);
        atom_add(o2 + 3, wx2 * y2d);
        atom_add(o2 + 4, wx2 * y2e);
    }
}

// ---------------------------------------------------------------------------
// Node kernel: WMMA f32 16x16x4 chained over K=32. One wave per 16-node tile.
// 9 components: comp0 uses lin_w0 (cols c), comps1-3 lin_w1 (cols 32+3c+i),
// comps4-8 lin_w2 (cols 128+5c+i). In-place on the msg buffer (== d_out):
// component column families are disjoint; within a component all loads happen
// before the dependent stores.
// ---------------------------------------------------------------------------
__global__ __launch_bounds__(128) void node_kernel(
    float* __restrict__ buf,          // in: msg ; out: final (in place)
    const float* __restrict__ lw0,
    const float* __restrict__ lw1,
    const float* __restrict__ lw2)
{
    const int tid  = threadIdx.x;
    const int wave = tid >> 5;
    const int lane = tid & 31;
    const int tile = blockIdx.x * 4 + wave;
    if (tile >= NTILES) return;           // wave-uniform: EXEC stays all-1s

    const int n0 = tile * 16;
    const int hi = lane >> 4;             // lane half (K interleave per ISA layout)
    const int mn = lane & 15;             // A row / B,D column-in-tile
    const float inv = 0.17677669529663687f;  // 1/sqrt(32)

    for (int comp = 0; comp < 9; ++comp) {
        const float* W;
        int cbase, istr, ioff;
        if (comp == 0)      { W = lw0; cbase = 0;   istr = 1; ioff = 0; }
        else if (comp < 4)  { W = lw1; cbase = 32;  istr = 3; ioff = comp - 1; }
        else                { W = lw2; cbase = 128; istr = 5; ioff = comp - 4; }

        // A fragments: 16x4 f32 per chunk. VGPR0 -> K = k0, VGPR1 -> K = k0+1
        // where k0 = 4*kc + 2*hi (lanes 16-31 carry K=+2 per ISA table).
        v2f Af[8];
        const long long rowbase = (long long)(n0 + mn) * OUTC + cbase + ioff;
#pragma unroll
        for (int kc = 0; kc < 8; ++kc) {
            const int k0 = kc * 4 + hi * 2;
            v2f a;
            a.x = buf[rowbase + (long long)k0 * istr];
            a.y = buf[rowbase + (long long)(k0 + 1) * istr];
            Af[kc] = a;
        }

        v8f acc[2];
#pragma unroll
        for (int nt = 0; nt < 2; ++nt) {
            const int ncol = nt * 16 + mn;
            v8f c = {0.f, 0.f, 0.f, 0.f, 0.f, 0.f, 0.f, 0.f};
#pragma unroll
            for (int kc = 0; kc < 8; ++kc) {
                const int k0 = kc * 4 + hi * 2;
                v2f b;
                b.x = W[k0 * CC + ncol];            // B 4x16: mirrored layout
                b.y = W[(k0 + 1) * CC + ncol];
                c = __builtin_amdgcn_wmma_f32_16x16x4_f32(
                        /*neg_a=*/false, Af[kc],
                        /*neg_b=*/false, b,
                        /*c_mod=*/(short)0, c,
                        /*reuse_a=*/false, /*reuse_b=*/false);
            }
            acc[nt] = c;
        }

        // Store after ALL reads of this component (in-place safety).
#pragma unroll
        for (int nt = 0; nt < 2; ++nt) {
            const int d = nt * 16 + mn;
            const long long ocol = (long long)cbase + (long long)d * istr + ioff;
#pragma unroll
            for (int v = 0; v < 8; ++v) {
                buf[(long long)(n0 + v + hi * 8) * OUTC + ocol] = acc[nt][v] * inv;
            }
        }
    }
}

extern "C" void kernel_launch(void* const* d_in, const int* in_sizes, int n_in,
                              void* d_out, int out_size, void* d_ws, size_t ws_size,
                              hipStream_t stream) {
    (void)in_sizes; (void)n_in; (void)out_size; (void)d_ws; (void)ws_size;

    const float*     lenght = (const float*)d_in[0];
    const float*     nodef  = (const float*)d_in[1];
    const float*     eattr  = (const float*)d_in[2];
    const long long* eidx   = (const long long*)d_in[3];   // int64 in reference
    const float*     fw0    = (const float*)d_in[4];
    const float*     fw1    = (const float*)d_in[5];
    const float*     fw2    = (const float*)d_in[6];
    const float*     fw3    = (const float*)d_in[7];
    const float*     lw0    = (const float*)d_in[8];
    const float*     lw1    = (const float*)d_in[9];
    const float*     lw2    = (const float*)d_in[10];
    float* out = (float*)d_out;

    const int total = NN * OUTC;
    zero_kernel<<<(total + 255) / 256, 256, 0, stream>>>(out, total);
    edge_kernel<<<(NE + WPB * EPW - 1) / (WPB * EPW), 128, 0, stream>>>(
        lenght, nodef, eattr, eidx, fw0, fw1, fw2, fw3, out);
    node_kernel<<<(NTILES + 3) / 4, 128, 0, stream>>>(out, lw0, lw1, lw2);
}